// QuantumFusion_53867479826801
// MI455X (gfx1250) — compile-verified
//
#include <hip/hip_runtime.h>
#include <math.h>

// ---------------------------------------------------------------------------
// QuantumFusion for MI455X (gfx1250, wave32)
//
// Roofline: feature reads = 16384*(768+512+1024)*4B = 151 MB -> ~6.5us at
// 23.3 TB/s. Streams features once; tall-skinny GEMMs (N=6 padded to 16) use
// V_WMMA_F32_16X16X4_F32 (exact fp32, matches the fp32 reference).
//
// Circuit reformulation: encoding gives a real product state s0[64]; the rest
// of the circuit is a fixed 64x64 unitary U(qw). Then
//   z_i = s0^T R_i s0,  R_i[a][b] = sum_j sign_i(j) Re(U[j,a] conj(U[j,b]))
// R (6*64*64 f32 = 98KB) is precomputed per launch and stays L2-resident.
//
// Parallel structure: 1024 blocks x 96 threads. Each block owns a 16-sample
// tile; wave m (of 3) runs modality m's GEMM (compile-time K, scalar control
// flow via readfirstlane), waves fuse angles through LDS, then each wave
// evaluates 2 of the 6 quadratic forms (a-dim split across the two 16-lane
// halves, combined with shfl_xor 16).
// ---------------------------------------------------------------------------

typedef __attribute__((ext_vector_type(2))) float v2f;
typedef __attribute__((ext_vector_type(8))) float v8f;

#define NQ 6
#define BATCH 16384
#define TPB 96  // 3 waves: one per modality
#define PI_F 3.14159265358979323846f

#define K0 768
#define K1 512
#define K2 1024
#define R_FLOATS (6 * 64 * 64)
#define WT_FLOATS (16 * (K0 + K1 + K2))

// ============================ setup kernel =================================
// 1 block, 64 threads. Thread t evolves basis column t of U in LDS, then all
// threads cooperatively emit R[6][64][64] to global workspace.
__global__ void qf_setup_kernel(const float* __restrict__ qw,  // [2][6][3]
                                float* __restrict__ R) {       // [6][64][64]
  __shared__ float2 U[64 * 64];  // U[col*64 + j] = amplitude j of column col
  const int col = threadIdx.x;   // 0..63
  float2* A = &U[col * 64];
  for (int j = 0; j < 64; ++j) A[j] = make_float2(j == col ? 1.f : 0.f, 0.f);

  // apply 1-qubit gate [[g00,g01],[g10,g11]] (complex) on wire w (w0 = MSB)
  auto ap1q = [&](int w, float g00r, float g00i, float g01r, float g01i,
                  float g10r, float g10i, float g11r, float g11i) {
    const int tb = 1 << (5 - w);
    const int low = tb - 1;
    for (int p = 0; p < 32; ++p) {
      int a0 = ((p & ~low) << 1) | (p & low);
      int a1 = a0 | tb;
      float2 x = A[a0], y = A[a1];
      A[a0] = make_float2(g00r * x.x - g00i * x.y + g01r * y.x - g01i * y.y,
                          g00r * x.y + g00i * x.x + g01r * y.y + g01i * y.x);
      A[a1] = make_float2(g10r * x.x - g10i * x.y + g11r * y.x - g11i * y.y,
                          g10r * x.y + g10i * x.x + g11r * y.y + g11i * y.x);
    }
  };
  auto cnot = [&](int cw, int tw) {
    const int cb = 1 << (5 - cw), tb = 1 << (5 - tw);
    for (int j = 0; j < 64; ++j)
      if ((j & cb) && !(j & tb)) {
        float2 tmp = A[j]; A[j] = A[j | tb]; A[j | tb] = tmp;
      }
  };

  for (int l = 0; l < 2; ++l) {
    for (int i = 0; i < NQ; ++i) {
      float tx = qw[(l * NQ + i) * 3 + 0];
      float ty = qw[(l * NQ + i) * 3 + 1];
      float tz = qw[(l * NQ + i) * 3 + 2];
      float cx = cosf(tx * 0.5f), sx = sinf(tx * 0.5f);
      ap1q(i, cx, 0.f, 0.f, -sx, 0.f, -sx, cx, 0.f);       // RX
      float cy = cosf(ty * 0.5f), sy = sinf(ty * 0.5f);
      ap1q(i, cy, 0.f, -sy, 0.f, sy, 0.f, cy, 0.f);        // RY
      float cz = cosf(tz * 0.5f), sz = sinf(tz * 0.5f);
      ap1q(i, cz, -sz, 0.f, 0.f, 0.f, 0.f, cz, sz);        // RZ
    }
    for (int i = 0; i < NQ - 1; ++i) cnot(i, i + 1);
    cnot(0, NQ - 1);
    for (int i = 0; i < NQ - 1; i += 2) {  // CRY(qw[l,i,1], i, i+1)
      float t = qw[(l * NQ + i) * 3 + 1];
      float c = cosf(t * 0.5f), s = sinf(t * 0.5f);
      const int cb = 1 << (5 - i), tb = 1 << (5 - (i + 1));
      for (int j = 0; j < 64; ++j)
        if ((j & cb) && !(j & tb)) {
          float2 x = A[j], y = A[j | tb];
          A[j]      = make_float2(c * x.x - s * y.x, c * x.y - s * y.y);
          A[j | tb] = make_float2(s * x.x + c * y.x, s * x.y + c * y.y);
        }
    }
  }
  __syncthreads();

  // R[i][a][b] = sum_j sign_i(j) * dot(U[:,a]_j, U[:,b]_j)
  for (int e = threadIdx.x; e < R_FLOATS; e += 64) {
    int i = e >> 12, a = (e >> 6) & 63, b = e & 63;
    const float2* Ua = &U[a * 64];
    const float2* Ub = &U[b * 64];
    float acc = 0.f;
    for (int j = 0; j < 64; ++j) {
      float d = Ua[j].x * Ub[j].x + Ua[j].y * Ub[j].y;
      acc += ((j >> (5 - i)) & 1) ? -d : d;
    }
    R[e] = acc;
  }
}

// ====================== W transpose-pad kernel =============================
// Wt layout (per modality, contiguous): Wt_m[n][k], n in [0,16), zeros for
// n >= 6. Makes the WMMA B-fragment a single contiguous global_load_b64.
__global__ void qf_pad_kernel(const float* __restrict__ W0,
                              const float* __restrict__ W1,
                              const float* __restrict__ W2,
                              float* __restrict__ Wt) {
  const int seg0 = 16 * K0, seg1 = seg0 + 16 * K1, total = WT_FLOATS;
  for (int e = blockIdx.x * blockDim.x + threadIdx.x; e < total;
       e += gridDim.x * blockDim.x) {
    const float* W;
    int idx, K;
    if (e < seg0)      { W = W0; idx = e;        K = K0; }
    else if (e < seg1) { W = W1; idx = e - seg0; K = K1; }
    else               { W = W2; idx = e - seg1; K = K2; }
    int n = idx / K, k = idx - n * K;
    Wt[e] = (n < NQ) ? W[k * NQ + n] : 0.f;
  }
}

// ============================ batch kernel =================================
// fp32 WMMA 16x16x4 fragment layouts (ISA 7.12.2):
//   A 16x4 (MxK): lanes 0-15 -> M=lane, VGPR{0,1}=K{0,1}; lanes 16-31 K{2,3}
//   B 4x16 (KxN): VGPR v: lanes 0-15 K=v, lanes 16-31 K=v+2; N=lane&15
//   C 16x16: VGPR v: M = v + 8*(lane>>4), N = lane&15
// K is a template parameter so the loop has a literal trip count: scalar
// loop control, unroll x4, immediate-offset b64 loads, EXEC stays all-ones.
template <bool PAD, int K>
__device__ inline v8f gemm16(const float* __restrict__ feat,
                             const float* __restrict__ Wsrc, int row0,
                             int lane15, int khalf) {
  v8f c = {0.f, 0.f, 0.f, 0.f, 0.f, 0.f, 0.f, 0.f};
  const float* arow = feat + (size_t)(row0 + lane15) * K + 2 * khalf;
  if (PAD) {
    // Wsrc = Wt_m[16][K]; B frag is one b64 load, no masking ALU
    const float* brow = Wsrc + (size_t)lane15 * K + 2 * khalf;
#pragma unroll 4
    for (int k0 = 0; k0 < K; k0 += 4) {
      v2f a = *reinterpret_cast<const v2f*>(arow + k0);
      v2f b = *reinterpret_cast<const v2f*>(brow + k0);
      c = __builtin_amdgcn_wmma_f32_16x16x4_f32(false, a, false, b, (short)0,
                                                c, false, false);
    }
  } else {
    const int nn = lane15 < NQ ? lane15 : 0;     // branchless N>=6 padding
    const float msk = lane15 < NQ ? 1.f : 0.f;   // keeps EXEC all-ones
#pragma unroll 4
    for (int k0 = 0; k0 < K; k0 += 4) {
      v2f a = *reinterpret_cast<const v2f*>(arow + k0);
      int kb = k0 + 2 * khalf;
      v2f b;
      b.x = msk * Wsrc[kb * NQ + nn];
      b.y = msk * Wsrc[(kb + 1) * NQ + nn];
      c = __builtin_amdgcn_wmma_f32_16x16x4_f32(false, a, false, b, (short)0,
                                                c, false, false);
    }
  }
  return c;
}

template <bool PAD>
__global__ void __launch_bounds__(TPB) qf_main_kernel(
    const float* __restrict__ feat0, const float* __restrict__ feat1,
    const float* __restrict__ feat2, const float* __restrict__ Wm0,
    const float* __restrict__ b0, const float* __restrict__ Wm1,
    const float* __restrict__ b1, const float* __restrict__ Wm2,
    const float* __restrict__ b2, const float* __restrict__ flog,
    const float* __restrict__ R, float* __restrict__ out) {
  __shared__ float angLDS[3][16][9];   // per-wave fused-angle contributions
  __shared__ float s0LDS[16][68];      // product state, pad 68: conflict-free

  const int lane = threadIdx.x & 31;
  // Wave id as an SGPR: modality / i-pair selection becomes pure SALU
  // control flow (s_cmp/s_cbranch), no EXEC masking around the WMMA loops.
  const int wvu = __builtin_amdgcn_readfirstlane(threadIdx.x >> 5);  // 0..2
  const int n = lane & 15;             // B column / C column / A row-in-tile
  const int khalf = lane >> 4;         // which K/M half this lane carries
  const int row0 = blockIdx.x * 16;    // one 16-sample tile per block

  // --- softmax fusion weights (3 scalars) ---
  float l0 = flog[0], l1 = flog[1], l2 = flog[2];
  float mx = fmaxf(l0, fmaxf(l1, l2));
  float e0 = expf(l0 - mx), e1 = expf(l1 - mx), e2 = expf(l2 - mx);
  float inv = 1.f / (e0 + e1 + e2);

  // --- this wave's modality GEMM (fp32 WMMA, compile-time K) ---
  v8f cfr;
  const float* bm;
  float fw;
  if (wvu == 0) {
    cfr = gemm16<PAD, K0>(feat0, Wm0, row0, n, khalf);
    bm = b0; fw = e0 * inv;
  } else if (wvu == 1) {
    cfr = gemm16<PAD, K1>(feat1, Wm1, row0, n, khalf);
    bm = b1; fw = e1 * inv;
  } else {
    cfr = gemm16<PAD, K2>(feat2, Wm2, row0, n, khalf);
    bm = b2; fw = e2 * inv;
  }

  float bb = (n < NQ) ? bm[n] : 0.f;

  // --- per-modality contribution fw * tanh(q), transposed via LDS ---
#pragma unroll
  for (int v = 0; v < 8; ++v) {
    float contrib = fw * tanhf(cfr[v] + bb);
    if (n < NQ) angLDS[wvu][v + 8 * khalf][n] = contrib;
  }
  __syncthreads();

  // --- build product state s0[64] for this lane's sample (smp = n) ---
  const int smp = n;
  float cw[6], sw[6];
#pragma unroll
  for (int w = 0; w < 6; ++w) {
    float a =
        PI_F * (angLDS[0][smp][w] + angLDS[1][smp][w] + angLDS[2][smp][w]) *
        0.5f;
    cw[w] = cosf(a);
    sw[w] = sinf(a);
  }
  float p[64];  // registers only (all indices static)
#pragma unroll
  for (int j = 0; j < 64; ++j) {
    float vv = ((j >> 5) & 1) ? sw[0] : cw[0];
    vv *= ((j >> 4) & 1) ? sw[1] : cw[1];
    vv *= ((j >> 3) & 1) ? sw[2] : cw[2];
    vv *= ((j >> 2) & 1) ? sw[3] : cw[3];
    vv *= ((j >> 1) & 1) ? sw[4] : cw[4];
    vv *= (j & 1) ? sw[5] : cw[5];
    p[j] = vv;
  }
  // stash s0 in LDS for the dynamically-indexed p[a] access (wave 0 writes)
  if (wvu == 0 && khalf == 0) {
#pragma unroll
    for (int j4 = 0; j4 < 16; ++j4) {
      float4 q = make_float4(p[4 * j4], p[4 * j4 + 1], p[4 * j4 + 2],
                             p[4 * j4 + 3]);
      *reinterpret_cast<float4*>(&s0LDS[smp][4 * j4]) = q;
    }
  }
  __syncthreads();

  // --- z_i = s0^T R_i s0 for this wave's two i's; a-half per lane-half ---
  const int abase = khalf * 32;
  const int i0 = wvu * 2;
  float z[2];
#pragma unroll
  for (int ii = 0; ii < 2; ++ii) {
    const int i = i0 + ii;
    float acc = 0.f;
    for (int aa = 0; aa < 32; ++aa) {
      int a = abase + aa;
      const float4* rrow =
          reinterpret_cast<const float4*>(R + ((i << 12) | (a << 6)));
      float t = 0.f;
#pragma unroll
      for (int b4 = 0; b4 < 16; ++b4) {  // uniform per half-wave: L0 hits
        float4 r4 = rrow[b4];
        t += r4.x * p[4 * b4] + r4.y * p[4 * b4 + 1] + r4.z * p[4 * b4 + 2] +
             r4.w * p[4 * b4 + 3];
      }
      acc = fmaf(s0LDS[smp][a], t, acc);
    }
    z[ii] = acc;
  }
  // combine the two a-halves (wave32 xor-16 shuffle)
#pragma unroll
  for (int ii = 0; ii < 2; ++ii) z[ii] += __shfl_xor(z[ii], 16, 32);

  if (khalf == 0) {
    out[(size_t)(row0 + smp) * NQ + i0 + 0] = z[0];
    out[(size_t)(row0 + smp) * NQ + i0 + 1] = z[1];
  }
}

// ============================ launcher =====================================
extern "C" void kernel_launch(void* const* d_in, const int* in_sizes, int n_in,
                              void* d_out, int out_size, void* d_ws,
                              size_t ws_size, hipStream_t stream) {
  // setup_inputs() dict order:
  // 0:feat0 1:W0 2:b0 3:feat1 4:W1 5:b1 6:feat2 7:W2 8:b2 9:qw 10:fusion_logits
  const float* feat0 = (const float*)d_in[0];
  const float* W0    = (const float*)d_in[1];
  const float* b0    = (const float*)d_in[2];
  const float* feat1 = (const float*)d_in[3];
  const float* W1    = (const float*)d_in[4];
  const float* b1    = (const float*)d_in[5];
  const float* feat2 = (const float*)d_in[6];
  const float* W2    = (const float*)d_in[7];
  const float* b2    = (const float*)d_in[8];
  const float* qw    = (const float*)d_in[9];
  const float* flog  = (const float*)d_in[10];
  float* out = (float*)d_out;

  float* R  = (float*)d_ws;                 // 98304 bytes
  float* Wt = R + R_FLOATS;                 // 147456 bytes (padded W copies)
  const size_t need = (size_t)(R_FLOATS + WT_FLOATS) * sizeof(float);

  qf_setup_kernel<<<1, 64, 0, stream>>>(qw, R);

  const int blocks = BATCH / 16;  // 1024 tiles, one per block (3 waves each)
  if (ws_size >= need) {
    qf_pad_kernel<<<144, 256, 0, stream>>>(W0, W1, W2, Wt);
    float* Wt0 = Wt;
    float* Wt1 = Wt0 + 16 * K0;
    float* Wt2 = Wt1 + 16 * K1;
    qf_main_kernel<true><<<blocks, TPB, 0, stream>>>(
        feat0, feat1, feat2, Wt0, b0, Wt1, b1, Wt2, b2, flog, R, out);
  } else {
    qf_main_kernel<false><<<blocks, TPB, 0, stream>>>(
        feat0, feat1, feat2, W0, b0, W1, b1, W2, b2, flog, R, out);
  }
}